// SparseDownsample_12077448036740
// MI455X (gfx1250) — compile-verified
//
#include <hip/hip_runtime.h>
#include <stdint.h>

#define CCH 128
#define HASH_BITS 20
#define HASH_SIZE (1u << HASH_BITS)
#define NBUCKET 65536
#define BUCKET_SHIFT 13
#define ROWS_PER_WAVE 4

// ---------------------------------------------------------------- utilities

__device__ __forceinline__ int hash_slot(int code) {
  unsigned h = (unsigned)code * 2654435761u;
  return (int)(h & (HASH_SIZE - 1u));
}

__device__ __forceinline__ void async_g2l_b128(unsigned ldsoff, const float* gsrc) {
  // per-lane: copies 16 bytes from global address gsrc to LDS offset ldsoff   (tracked by ASYNCcnt)
  asm volatile("global_load_async_to_lds_b128 %0, %1, off"
               :: "v"(ldsoff), "v"(gsrc) : "memory");
}

__device__ __forceinline__ void async_l2g_b128(float* gdst, unsigned ldsoff) {
  // per-lane: copies 16 bytes from LDS offset ldsoff to global address gdst
  asm volatile("global_store_async_from_lds_b128 %0, %1, off"
               :: "v"(gdst), "v"(ldsoff) : "memory");
}

__device__ __forceinline__ void wait_async0() {
#if __has_builtin(__builtin_amdgcn_s_wait_asynccnt)
  __builtin_amdgcn_s_wait_asynccnt(0);
#else
  asm volatile("s_wait_asynccnt 0" ::: "memory");
#endif
}

// ---------------------------------------------------------------- hot kernel
// One wave per output row group. Unique voxels (99.95% of rows): rows are
// moved with the CDNA5 async data mover, 4 rows in flight per wave
// (4 x 512B = 2KB MLP), a single s_wait_asynccnt 0 between the load batch and
// the store batch (at that point only loads are outstanding, so the wait is
// exact). Store completion is covered by S_ENDPGM's implicit wait-idle.
// Duplicate voxels: float atomics into the pre-zeroed row.
__global__ void __launch_bounds__(256) k_copy(const float* __restrict__ feats,
                                              const int* __restrict__ rank,
                                              const int* __restrict__ counts,
                                              float* __restrict__ outf, int n) {
  __shared__ float smem[8 * ROWS_PER_WAVE * CCH];    // 8 waves * 4 rows * 512B = 16KB
  int wave = threadIdx.x >> 5;
  int lane = threadIdx.x & 31;
  int base = (blockIdx.x * 8 + wave) * ROWS_PER_WAVE;

  unsigned pend = 0;                                 // rows with an async load in flight
  int rr[ROWS_PER_WAVE];

  // phase 1: issue async loads (unique rows) / atomic accumulate (dup rows)
  for (int k = 0; k < ROWS_PER_WAVE; ++k) {
    int i = base + k;
    if (i >= n) break;
    int r = rank[i];                                 // wave-uniform
    rr[k] = r;
    const float* src = feats + (size_t)i * CCH + lane * 4;
    if (counts[r] == 1) {
      unsigned ldsoff =
          (unsigned)(uintptr_t)(&smem[(wave * ROWS_PER_WAVE + k) * CCH + lane * 4]);
      async_g2l_b128(ldsoff, src);                   // wave moves 32 x 16B = 512B
      pend |= 1u << k;
    } else {
      float4 v = *(const float4*)src;
      float* dst = outf + (size_t)r * CCH + lane * 4;
      atomicAdd(dst + 0, v.x);
      atomicAdd(dst + 1, v.y);
      atomicAdd(dst + 2, v.z);
      atomicAdd(dst + 3, v.w);
    }
  }

  if (pend == 0) return;
  wait_async0();                                     // all row loads landed in LDS

  // phase 2: issue async stores; S_ENDPGM waits for their completion
  for (int k = 0; k < ROWS_PER_WAVE; ++k) {
    if (pend & (1u << k)) {
      unsigned ldsoff =
          (unsigned)(uintptr_t)(&smem[(wave * ROWS_PER_WAVE + k) * CCH + lane * 4]);
      float* dst = outf + (size_t)rr[k] * CCH + lane * 4;
      async_l2g_b128(dst, ldsoff);
    }
  }
}

// ---------------------------------------------------------------- pipeline

// re-initialize all scratch state (harness does not re-poison between replays)
__global__ void k_init(int* hk, int* bc, int* bf, int* counts, int* uniq_n, int n) {
  int i = blockIdx.x * blockDim.x + threadIdx.x;
  int stride = gridDim.x * blockDim.x;
  for (int s = i; s < (int)HASH_SIZE; s += stride) hk[s] = -1;
  for (int s = i; s < NBUCKET; s += stride) { bc[s] = 0; bf[s] = 0; }
  for (int s = i; s < n; s += stride) counts[s] = 0;
  if (i == 0) *uniq_n = 0;
}

// code per point + hash-insert; first inserter of a code bumps its bucket count
__global__ void k_code_insert(const int* __restrict__ coords, int* __restrict__ codes,
                              int* __restrict__ hk, int* __restrict__ bc, int n) {
  int i = blockIdx.x * blockDim.x + threadIdx.x;
  if (i >= n) return;
  int b = coords[4 * i + 0];
  int x = coords[4 * i + 1] >> 1;   // // FACTOR (values are non-negative)
  int y = coords[4 * i + 2] >> 1;
  int z = coords[4 * i + 3] >> 1;
  int code = (b << 21) + (x << 14) + (y << 7) + z;    // < 2^29, fits int32
  codes[i] = code;
  int s = hash_slot(code);
  for (;;) {
    int old = atomicCAS(&hk[s], -1, code);
    if (old == -1) { atomicAdd(&bc[code >> BUCKET_SHIFT], 1); break; }
    if (old == code) break;
    s = (s + 1) & (int)(HASH_SIZE - 1u);
  }
}

// exclusive prefix over 65536 bucket counts, single 1024-thread block (LDS scan)
__global__ void __launch_bounds__(1024) k_scan(const int* __restrict__ bc,
                                               int* __restrict__ boff,
                                               int* __restrict__ uniq_n) {
  __shared__ int sdata[1024];
  int t = threadIdx.x;
  int base = t * 64;
  int sum = 0;
  for (int k = 0; k < 64; ++k) sum += bc[base + k];
  sdata[t] = sum;
  __syncthreads();
  for (int off = 1; off < 1024; off <<= 1) {
    int v = (t >= off) ? sdata[t - off] : 0;
    __syncthreads();
    sdata[t] += v;
    __syncthreads();
  }
  int run = sdata[t] - sum;                 // exclusive prefix for this thread
  for (int k = 0; k < 64; ++k) { boff[base + k] = run; run += bc[base + k]; }
  if (t == 1023) *uniq_n = sdata[1023];     // total number of unique codes U
}

// compact filled hash slots into bucket-major segments
__global__ void k_scatter(const int* __restrict__ hk, const int* __restrict__ boff,
                          int* __restrict__ bf, int* __restrict__ sorted) {
  int i = blockIdx.x * blockDim.x + threadIdx.x;
  int stride = gridDim.x * blockDim.x;
  for (int s = i; s < (int)HASH_SIZE; s += stride) {
    int c = hk[s];
    if (c != -1) {
      int b = c >> BUCKET_SHIFT;
      int p = boff[b] + atomicAdd(&bf[b], 1);
      sorted[p] = c;
    }
  }
}

// tiny per-bucket insertion sort -> 'sorted' is globally ascending
// (bucket = high bits of code, so bucket-major + in-bucket ascending == sorted)
__global__ void k_bucket_sort(const int* __restrict__ boff, const int* __restrict__ bc,
                              int* __restrict__ sorted) {
  int i = blockIdx.x * blockDim.x + threadIdx.x;
  int stride = gridDim.x * blockDim.x;
  for (int b = i; b < NBUCKET; b += stride) {
    int st = boff[b], cnt = bc[b];
    for (int j = 1; j < cnt; ++j) {
      int key = sorted[st + j];
      int k = j - 1;
      while (k >= 0 && sorted[st + k] > key) { sorted[st + k + 1] = sorted[st + k]; --k; }
      sorted[st + k + 1] = key;
    }
  }
}

// rank (== jnp.unique inverse index) per point + duplicate counts
__global__ void k_rank(const int* __restrict__ codes, const int* __restrict__ boff,
                       const int* __restrict__ bc, const int* __restrict__ sorted,
                       int* __restrict__ rank, int* __restrict__ counts, int n) {
  int i = blockIdx.x * blockDim.x + threadIdx.x;
  if (i >= n) return;
  int code = codes[i];
  int b = code >> BUCKET_SHIFT;
  int lo = boff[b], hi = lo + bc[b] - 1, r = lo;
  while (lo <= hi) {
    int mid = (lo + hi) >> 1;
    int v = sorted[mid];
    if (v < code) lo = mid + 1;
    else if (v > code) hi = mid - 1;
    else { r = mid; break; }
  }
  rank[i] = r;
  atomicAdd(&counts[r], 1);
}

// decode coords for every output row; zero only contested + padded feature rows
__global__ void k_rows(const int* __restrict__ sorted, const int* __restrict__ counts,
                       const int* __restrict__ uniq_n, float* __restrict__ outf,
                       int* __restrict__ outc, int n) {
  int i = blockIdx.x * blockDim.x + threadIdx.x;
  int stride = gridDim.x * blockDim.x;
  int U = *uniq_n;
  for (int r = i; r < n; r += stride) {
    bool zero_row;
    if (r < U) {
      int code = sorted[r];
      outc[4 * r + 0] = code >> 21;         // code >= 0
      outc[4 * r + 1] = (code >> 14) & 127;
      outc[4 * r + 2] = (code >> 7) & 127;
      outc[4 * r + 3] = code & 127;
      zero_row = (counts[r] > 1);
    } else {
      // uniq == -1 padding: jnp floor-div/mod on -1 gives [-1,127,127,127]
      outc[4 * r + 0] = -1;
      outc[4 * r + 1] = 127;
      outc[4 * r + 2] = 127;
      outc[4 * r + 3] = 127;
      zero_row = true;
    }
    if (zero_row) {
      float4 z = make_float4(0.f, 0.f, 0.f, 0.f);
      float4* row = (float4*)(outf + (size_t)r * CCH);
      for (int c = 0; c < CCH / 4; ++c) row[c] = z;
    }
  }
}

// mean for the rare contested rows
__global__ void k_div(const int* __restrict__ counts, const int* __restrict__ uniq_n,
                      float* __restrict__ outf, int n) {
  int i = blockIdx.x * blockDim.x + threadIdx.x;
  int stride = gridDim.x * blockDim.x;
  int U = *uniq_n;
  for (int r = i; r < n; r += stride) {
    if (r < U && counts[r] > 1) {
      float w = 1.0f / (float)counts[r];
      float* row = outf + (size_t)r * CCH;
      for (int c = 0; c < CCH; ++c) row[c] *= w;
    }
  }
}

// ---------------------------------------------------------------- launcher

extern "C" void kernel_launch(void* const* d_in, const int* in_sizes, int n_in,
                              void* d_out, int out_size, void* d_ws, size_t ws_size,
                              hipStream_t stream) {
  const float* feats = (const float*)d_in[0];
  const int* coords = (const int*)d_in[1];
  int n = in_sizes[0] / CCH;                     // N active voxels

  float* outf = (float*)d_out;                   // [N, 128] means
  int* outc = (int*)(outf + (size_t)n * CCH);    // [N, 4] coords (int32 bits)

  // workspace carve-up (~12.8 MB)
  char* w = (char*)d_ws;
  int* hk     = (int*)w; w += (size_t)HASH_SIZE * 4;
  int* bc     = (int*)w; w += (size_t)NBUCKET * 4;
  int* boff   = (int*)w; w += (size_t)NBUCKET * 4;
  int* bf     = (int*)w; w += (size_t)NBUCKET * 4;
  int* codes  = (int*)w; w += (size_t)n * 4;
  int* rank   = (int*)w; w += (size_t)n * 4;
  int* sorted = (int*)w; w += (size_t)n * 4;
  int* counts = (int*)w; w += (size_t)n * 4;
  int* uniq_n = (int*)w;

  const int B = 256;
  int gridN = (n + B - 1) / B;
  int gridH = (int)((HASH_SIZE + B - 1) / B);
  int rowsPerBlock = 8 * ROWS_PER_WAVE;          // 8 waves/block * 4 rows/wave

  k_init<<<gridH, B, 0, stream>>>(hk, bc, bf, counts, uniq_n, n);
  k_code_insert<<<gridN, B, 0, stream>>>(coords, codes, hk, bc, n);
  k_scan<<<1, 1024, 0, stream>>>(bc, boff, uniq_n);
  k_scatter<<<gridH, B, 0, stream>>>(hk, boff, bf, sorted);
  k_bucket_sort<<<NBUCKET / B, B, 0, stream>>>(boff, bc, sorted);
  k_rank<<<gridN, B, 0, stream>>>(codes, boff, bc, sorted, rank, counts, n);
  k_rows<<<gridN, B, 0, stream>>>(sorted, counts, uniq_n, outf, outc, n);
  k_copy<<<(n + rowsPerBlock - 1) / rowsPerBlock, B, 0, stream>>>(feats, rank, counts, outf, n);
  k_div<<<gridN, B, 0, stream>>>(counts, uniq_n, outf, n);
}